// PhysicsInformedLoss_59760174957349
// MI455X (gfx1250) — compile-verified
//
#include <hip/hip_runtime.h>
#include <hip/hip_bf16.h>

// Physics-informed loss, fused single pass.
// B=32, C=3, H=W=512, fp32. Memory-bound (AI ~2 flop/byte): target is the
// 23.3 TB/s HBM roofline via float4 loads + single-pass fusion; reductions
// use V_WMMA_F32_16X16X4_F32 as the wave-level sum engine (CDNA5 matrix path).

#define HH 512
#define WW 512
#define HWSZ (HH * WW)

constexpr float INV_DX  = 100.0f;     // 1/0.01
constexpr float INV_DX2 = 10000.0f;   // 1/0.0001
constexpr float INV_DT  = 1000.0f;    // 1/0.001

typedef float v2f __attribute__((ext_vector_type(2)));
typedef float v8f __attribute__((ext_vector_type(8)));

// ---- 32-lane sum via WMMA f32 16x16x4 ----------------------------------
// A (16x4): VGPR0 -> lanes0-15 = A[m,0], lanes16-31 = A[m,2]; VGPR1 = 0.
// B (4x16): all ones.  D[m,n] = A[m,0]+A[m,2] = x[m] + x[m+16] (any n).
// D VGPR r: lanes0-15 hold M=r, lanes16-31 hold M=8+r, so summing the 8 D
// VGPRs gives half-sums in each 16-lane group; one xor-16 shuffle finishes.
// EXEC must be all ones at the call site (full 256-thread blocks, no
// divergence at the call).
__device__ __forceinline__ float wave_sum32(float x) {
    v2f a;  a[0] = x;    a[1] = 0.0f;
    v2f bo; bo[0] = 1.0f; bo[1] = 1.0f;
    v8f c = {};
    v8f d = __builtin_amdgcn_wmma_f32_16x16x4_f32(
        /*neg_a=*/false, a, /*neg_b=*/false, bo,
        /*c_mod=*/(short)0, c, /*reuse_a=*/false, /*reuse_b=*/false);
    float s = ((d[0] + d[1]) + (d[2] + d[3])) + ((d[4] + d[5]) + (d[6] + d[7]));
    s += __shfl_xor(s, 16);
    return s;  // every lane holds the full 32-lane sum
}

__device__ __forceinline__ void load8(float* dst, const float* __restrict__ src) {
    float4 a = *(const float4*)(src);
    float4 b = *(const float4*)(src + 4);
    dst[0] = a.x; dst[1] = a.y; dst[2] = a.z; dst[3] = a.w;
    dst[4] = b.x; dst[5] = b.y; dst[6] = b.z; dst[7] = b.w;
}
__device__ __forceinline__ void zero8(float* dst) {
#pragma unroll
    for (int i = 0; i < 8; ++i) dst[i] = 0.0f;
}

// Kernel 1: grid (B, H/4), 256 threads. Each thread: 8 pixels of one row
// (two float4 loads per row segment, 16B aligned since W=512).
// Writes 3 partial sums per block to ws (no atomics -> deterministic).
__global__ __launch_bounds__(256)
void pil_partial(const float* __restrict__ gen, const float* __restrict__ hr,
                 float* __restrict__ part) {
    const int b   = blockIdx.x;
    const int yg  = blockIdx.y;              // 0..127
    const int tid = threadIdx.x;             // 0..255
    const int y   = yg * 4 + (tid >> 6);     // 4 rows per block
    const int x0  = (tid & 63) * 8;          // 8 px per thread

    const float* Gu = gen + (size_t)b * 3 * HWSZ + (size_t)y * WW;
    const float* Gv = Gu + HWSZ;
    const float* Gp = Gv + HWSZ;
    const float* Hu = hr + (size_t)b * 3 * HWSZ + (size_t)y * WW;
    const float* Hv = Hu + HWSZ;

    const bool has_ym = (y > 0);
    const bool has_yp = (y < HH - 1);
    const bool has_xm = (x0 > 0);
    const bool has_xp = (x0 + 8 < WW);

    float uc[9], uym[8], vc[8], vym[8];
    float pc[8], pym[8], pyp[8];
    float huc[9], huym[8], hvc[9], hvym[8];

    load8(uc, Gu + x0);  uc[8]  = has_xp ? Gu[x0 + 8] : 0.0f;
    if (has_ym) load8(uym, Gu + x0 - WW); else zero8(uym);
    load8(vc, Gv + x0);
    if (has_ym) load8(vym, Gv + x0 - WW); else zero8(vym);
    load8(pc, Gp + x0);
    const float pm1 = has_xm ? Gp[x0 - 1] : 0.0f;
    const float pp8 = has_xp ? Gp[x0 + 8] : 0.0f;
    if (has_ym) load8(pym, Gp + x0 - WW); else zero8(pym);
    if (has_yp) load8(pyp, Gp + x0 + WW); else zero8(pyp);
    load8(huc, Hu + x0); huc[8] = has_xp ? Hu[x0 + 8] : 0.0f;
    if (has_ym) load8(huym, Hu + x0 - WW); else zero8(huym);
    load8(hvc, Hv + x0); hvc[8] = has_xp ? Hv[x0 + 8] : 0.0f;
    if (has_ym) load8(hvym, Hv + x0 - WW); else zero8(hvym);

    float s_srm = 0.0f, s_con = 0.0f, s_press = 0.0f;
#pragma unroll
    for (int i = 0; i < 8; ++i) {
        const float pxm = (i == 0) ? pm1 : pc[i - 1];
        const float pxp = (i == 7) ? pp8 : pc[i + 1];
        // gen derivatives (ddx forward w/ zero right-pad; ddy = (x[y-1]-x[y])/dx)
        const float dudx = (uc[i + 1] - uc[i]) * INV_DX;
        const float dvdy = (vym[i] - vc[i]) * INV_DX;
        const float con  = dudx + dvdy;
        const float d2px = (pxp - 2.0f * pc[i] + pxm) * INV_DX2;
        const float d2py = (pyp[i] - 2.0f * pc[i] + pym[i]) * INV_DX2;
        const float press = d2px * d2px + d2py * d2py
                          + dudx * dudx + dvdy * dvdy + 2.0f * dudx * dvdy
                          - con * INV_DT;
        // HR strain-rate magnitude
        const float hdudx = (huc[i + 1] - huc[i]) * INV_DX;
        const float hdvdy = (hvym[i] - hvc[i]) * INV_DX;
        const float hdudy = (huym[i] - huc[i]) * INV_DX;
        const float hdvdx = (hvc[i + 1] - hvc[i]) * INV_DX;
        const float sh    = 0.5f * (hdudy + hdvdx);
        const float srm   = hdudx * hdudx + hdvdy * hdvdy + 2.0f * sh * sh;

        s_srm   += srm;
        s_con   += con * con;
        s_press += press * press;
    }

    // wave-level (32-lane) reduction on the matrix pipe
    const float w_srm   = wave_sum32(s_srm);
    const float w_con   = wave_sum32(s_con);
    const float w_press = wave_sum32(s_press);

    __shared__ float sm[8][3];
    const int wave = tid >> 5;
    if ((tid & 31) == 0) {
        sm[wave][0] = w_srm; sm[wave][1] = w_con; sm[wave][2] = w_press;
    }
    __syncthreads();
    if (tid == 0) {
        float a0 = 0.0f, a1 = 0.0f, a2 = 0.0f;
#pragma unroll
        for (int w = 0; w < 8; ++w) { a0 += sm[w][0]; a1 += sm[w][1]; a2 += sm[w][2]; }
        const size_t o = ((size_t)b * gridDim.y + yg) * 3;
        part[o + 0] = a0; part[o + 1] = a1; part[o + 2] = a2;
    }
}

// Kernel 2: one block per batch; deterministic LDS tree over 128 partials,
// then the scalar loss formula.
__global__ __launch_bounds__(128)
void pil_final(const float* __restrict__ part, float* __restrict__ out) {
    const int b = blockIdx.x;
    const int t = threadIdx.x;  // 0..127
    __shared__ float sm0[128], sm1[128], sm2[128];
    const size_t o = ((size_t)b * 128 + t) * 3;
    sm0[t] = part[o + 0]; sm1[t] = part[o + 1]; sm2[t] = part[o + 2];
    __syncthreads();
#pragma unroll
    for (int s = 64; s > 0; s >>= 1) {
        if (t < s) { sm0[t] += sm0[t + s]; sm1[t] += sm1[t + s]; sm2[t] += sm2[t + s]; }
        __syncthreads();
    }
    if (t == 0) {
        const float mean = sm0[0] * (1.0f / (float)HWSZ);   // strain_rate_2_HR
        const float loss = 0.5f * sm1[0] / mean + 0.5f * sm2[0] / (mean * mean);
        out[b] = loss;
    }
}

extern "C" void kernel_launch(void* const* d_in, const int* in_sizes, int n_in,
                              void* d_out, int out_size, void* d_ws, size_t ws_size,
                              hipStream_t stream) {
    const float* gen = (const float*)d_in[0];
    const float* hr  = (const float*)d_in[1];
    float* out  = (float*)d_out;
    float* part = (float*)d_ws;                 // B * 128 * 3 floats

    const int B = in_sizes[0] / (3 * HWSZ);     // 32
    dim3 grid1((unsigned)B, HH / 4);
    pil_partial<<<grid1, 256, 0, stream>>>(gen, hr, part);
    pil_final<<<B, 128, 0, stream>>>(part, out);
}